// AdaptiveMultiAdaAttN_v2_17360257810905
// MI455X (gfx1250) — compile-verified
//
#include <hip/hip_runtime.h>
#include <math.h>

typedef float v2f __attribute__((ext_vector_type(2)));
typedef float v8f __attribute__((ext_vector_type(8)));

#define BATCH 4
#define CCH   256
#define QKC   512
#define NC    4096   // 64*64 content positions
#define NS    4096   // 64*64 resized-style positions

// jax.image.resize bicubic (a=-0.5), antialias=True, scale=1/2:
// taps at d = ±0.25,±0.75,±1.25,±1.75 (output units), x0.5, normalized.
__device__ __constant__ float RW[8] = {
    -0.01171875f, -0.03515625f, 0.11328125f, 0.43359375f,
     0.43359375f,  0.11328125f, -0.03515625f, -0.01171875f };

__device__ __forceinline__ v8f wmma_f32(v2f a, v2f b, v8f c) {
  // D = A(16x4 f32) * B(4x16 f32) + C(16x16 f32)
  return __builtin_amdgcn_wmma_f32_16x16x4_f32(false, a, false, b, (short)0, c,
                                               false, false);
}

// ---------------- separable bicubic downsample 128 -> 64 ----------------
__global__ void resize_h(const float* __restrict__ in, float* __restrict__ out) {
  // in: [B*C,128,128] -> out: [B*C,128,64]
  long idx = (long)blockIdx.x * blockDim.x + threadIdx.x;
  int  xo   = idx & 63;
  long rest = idx >> 6;
  int  y    = rest & 127;
  long pc   = rest >> 7;
  const float* row = in + (pc * 128 + y) * 128;
  float acc = 0.f, wsum = 0.f;
  int j0 = 2 * xo - 3;
#pragma unroll
  for (int t = 0; t < 8; ++t) {
    int j = j0 + t;
    if (j >= 0 && j < 128) { acc += RW[t] * row[j]; wsum += RW[t]; }
  }
  out[idx] = acc / wsum;
}

__global__ void resize_v(const float* __restrict__ in, float* __restrict__ out) {
  // in: [B*C,128,64] -> out: [B*C,64,64]
  long idx = (long)blockIdx.x * blockDim.x + threadIdx.x;
  int  x    = idx & 63;
  long rest = idx >> 6;
  int  yo   = rest & 63;
  long pc   = rest >> 6;
  const float* col = in + pc * (128 * 64) + x;
  float acc = 0.f, wsum = 0.f;
  int j0 = 2 * yo - 3;
#pragma unroll
  for (int t = 0; t < 8; ++t) {
    int j = j0 + t;
    if (j >= 0 && j < 128) { acc += RW[t] * col[(long)j * 64]; wsum += RW[t]; }
  }
  out[idx] = acc / wsum;
}

// ---------------- conv1x1 GEMM: Y[b,m,n] = sum_k W[m,k] X[b,k,n] + bias[m] --
// block = 256 (8 waves); wave -> 16x64 tile; WG -> 128x64 tile.
__global__ void conv1x1_wmma(const float* __restrict__ W, const float* __restrict__ X,
                             const float* __restrict__ bias, float* __restrict__ Y,
                             float* __restrict__ Ysq, int M, int K, int N) {
  int lane = threadIdx.x & 31;
  int wv   = threadIdx.x >> 5;
  int laneM = lane & 15;
  int laneHalf = lane >> 4;
  int kOff = laneHalf * 2;
  int m0 = blockIdx.y * 128 + wv * 16;
  int n0 = blockIdx.x * 64;
  int b  = blockIdx.z;
  const float* Xb = X + (long)b * K * N;
  const float* Wrow = W + (long)(m0 + laneM) * K + kOff;
  v8f acc[4] = {v8f{}, v8f{}, v8f{}, v8f{}};
  for (int k = 0; k < K; k += 4) {
    v2f a = *(const v2f*)(Wrow + k);                 // A[laneM][k+kOff..+1]
    const float* xk0 = Xb + (long)(k + kOff) * N + n0 + laneM;
    const float* xk1 = xk0 + N;
    v2f b0 = { xk0[0],  xk1[0]  };
    v2f b1 = { xk0[16], xk1[16] };
    v2f b2 = { xk0[32], xk1[32] };
    v2f b3 = { xk0[48], xk1[48] };
    acc[0] = wmma_f32(a, b0, acc[0]);
    acc[1] = wmma_f32(a, b1, acc[1]);
    acc[2] = wmma_f32(a, b2, acc[2]);
    acc[3] = wmma_f32(a, b3, acc[3]);
  }
  long base = (long)b * M * N;
#pragma unroll
  for (int t = 0; t < 4; ++t) {
#pragma unroll
    for (int r = 0; r < 8; ++r) {
      int m = m0 + r + 8 * laneHalf;
      int n = n0 + t * 16 + laneM;
      float v = acc[t][r] + bias[m];
      long o = base + (long)m * N + n;
      Y[o] = v;
      if (Ysq) Ysq[o] = v * v;
    }
  }
}

// ---------------- logits: L[b,n,m] = sum_k Fq[b,k,n] * Gk[b,k,m] ----------
__global__ void attn_logits_wmma(const float* __restrict__ Fq,
                                 const float* __restrict__ Gk,
                                 float* __restrict__ L) {
  int lane = threadIdx.x & 31;
  int wv   = threadIdx.x >> 5;
  int laneM = lane & 15;
  int laneHalf = lane >> 4;
  int kOff = laneHalf * 2;
  int n0 = blockIdx.y * 128 + wv * 16;   // content rows
  int m0 = blockIdx.x * 64;              // style cols
  int b  = blockIdx.z;
  const float* F = Fq + (long)b * QKC * NC;
  const float* G = Gk + (long)b * QKC * NS;
  v8f acc[4] = {v8f{}, v8f{}, v8f{}, v8f{}};
  for (int k = 0; k < QKC; k += 4) {
    const float* f0 = F + (long)(k + kOff) * NC + n0 + laneM;
    v2f a = { f0[0], f0[NC] };                       // A = Fq^T fragment
    const float* g0 = G + (long)(k + kOff) * NS + m0 + laneM;
    v2f b0 = { g0[0],  g0[NS]      };
    v2f b1 = { g0[16], g0[NS + 16] };
    v2f b2 = { g0[32], g0[NS + 32] };
    v2f b3 = { g0[48], g0[NS + 48] };
    acc[0] = wmma_f32(a, b0, acc[0]);
    acc[1] = wmma_f32(a, b1, acc[1]);
    acc[2] = wmma_f32(a, b2, acc[2]);
    acc[3] = wmma_f32(a, b3, acc[3]);
  }
  long base = (long)b * NC * NS;
#pragma unroll
  for (int t = 0; t < 4; ++t) {
#pragma unroll
    for (int r = 0; r < 8; ++r) {
      int n = n0 + r + 8 * laneHalf;
      int m = m0 + t * 16 + laneM;
      L[base + (long)n * NS + m] = acc[t][r];
    }
  }
}

// ---------------- per-row softmax stats (wave32 per row) ------------------
__global__ void row_softmax_stats(const float* __restrict__ L,
                                  float* __restrict__ rmax, float* __restrict__ rinv) {
  int lane = threadIdx.x & 31;
  int wv   = threadIdx.x >> 5;
  long row = (long)blockIdx.x * 4 + wv;          // 0..B*NC-1
  const float* p = L + row * NS;
  float mx = -1e30f;
  for (int i = lane; i < NS; i += 32) mx = fmaxf(mx, p[i]);
  for (int off = 16; off > 0; off >>= 1) mx = fmaxf(mx, __shfl_xor(mx, off, 32));
  float s = 0.f;
  for (int i = lane; i < NS; i += 32) s += __expf(p[i] - mx);
  for (int off = 16; off > 0; off >>= 1) s += __shfl_xor(s, off, 32);
  if (lane == 0) { rmax[row] = mx; rinv[row] = 1.f / s; }
}

// ---------------- content per-(b,c) mean / inv-std (unbiased, +EPS) -------
__global__ void content_stats(const float* __restrict__ X,
                              float* __restrict__ cm, float* __restrict__ civ) {
  int lane = threadIdx.x & 31;
  int wv   = threadIdx.x >> 5;
  long row = (long)blockIdx.x * 4 + wv;          // 0..B*C-1
  const float* p = X + row * NC;
  float s = 0.f, ss = 0.f;
  for (int i = lane; i < NC; i += 32) { float v = p[i]; s += v; ss += v * v; }
  for (int off = 16; off > 0; off >>= 1) { s += __shfl_xor(s, off, 32); ss += __shfl_xor(ss, off, 32); }
  if (lane == 0) {
    float m   = s / (float)NC;
    float var = (ss - (float)NC * m * m) / (float)(NC - 1) + 1e-5f;
    cm[row] = m; civ[row] = rsqrtf(var);
  }
}

// ------- mean = S@Hr^T, ex2 = S@(Hr^2)^T, fused std*mvn(content)+mean -----
// block = 256 (8 waves); WG -> 16 content rows x 256 channels; wave -> 32 ch.
__global__ void attn_apply_wmma(const float* __restrict__ L,
                                const float* __restrict__ rmax,
                                const float* __restrict__ rinv,
                                const float* __restrict__ Hr,
                                const float* __restrict__ Hr2,
                                const float* __restrict__ content,
                                const float* __restrict__ cm,
                                const float* __restrict__ civ,
                                float* __restrict__ out) {
  int lane = threadIdx.x & 31;
  int wv   = threadIdx.x >> 5;
  int laneM = lane & 15;
  int laneHalf = lane >> 4;
  int kOff = laneHalf * 2;
  int n0 = blockIdx.x * 16;
  int b  = blockIdx.y;
  int c0 = wv * 32;
  int nA = n0 + laneM;
  float mxA = rmax[(long)b * NC + nA];
  float ivA = rinv[(long)b * NC + nA];
  const float* lrow = L + ((long)b * NC + nA) * NS + kOff;
  const float* Hb   = Hr  + (long)b * CCH * NS;
  const float* H2b  = Hr2 + (long)b * CCH * NS;
  const float* h0 = Hb  + (long)(c0 + laneM)      * NS + kOff;
  const float* h1 = Hb  + (long)(c0 + 16 + laneM) * NS + kOff;
  const float* e0 = H2b + (long)(c0 + laneM)      * NS + kOff;
  const float* e1 = H2b + (long)(c0 + 16 + laneM) * NS + kOff;
  v8f aM0 = {}, aM1 = {}, aE0 = {}, aE1 = {};
  for (int m = 0; m < NS; m += 4) {
    v2f l  = *(const v2f*)(lrow + m);
    v2f a  = { __expf(l.x - mxA) * ivA, __expf(l.y - mxA) * ivA };  // S fragment
    v2f b0 = *(const v2f*)(h0 + m);
    v2f b1 = *(const v2f*)(h1 + m);
    v2f q0 = *(const v2f*)(e0 + m);
    v2f q1 = *(const v2f*)(e1 + m);
    aM0 = wmma_f32(a, b0, aM0);
    aM1 = wmma_f32(a, b1, aM1);
    aE0 = wmma_f32(a, q0, aE0);
    aE1 = wmma_f32(a, q1, aE1);
  }
#pragma unroll
  for (int t = 0; t < 2; ++t) {
    v8f am = t ? aM1 : aM0;
    v8f ae = t ? aE1 : aE0;
    int c = c0 + t * 16 + laneM;
    long bc = (long)b * CCH + c;
    float cmu = cm[bc], cis = civ[bc];
#pragma unroll
    for (int r = 0; r < 8; ++r) {
      int n = n0 + r + 8 * laneHalf;
      float mean = am[r];
      float ex2  = ae[r];
      float sd   = sqrtf(fmaxf(ex2 - mean * mean, 0.f));
      float x    = content[bc * NC + n];
      out[bc * NC + n] = sd * ((x - cmu) * cis) + mean;
    }
  }
}

// ---------------- coalesced S output --------------------------------------
__global__ void write_S(const float* __restrict__ L, const float* __restrict__ rmax,
                        const float* __restrict__ rinv, float* __restrict__ S) {
  long idx = (long)blockIdx.x * 256 + threadIdx.x;
  long row = idx >> 12;   // / NS
  S[idx] = __expf(L[idx] - rmax[row]) * rinv[row];
}

extern "C" void kernel_launch(void* const* d_in, const int* in_sizes, int n_in,
                              void* d_out, int out_size, void* d_ws, size_t ws_size,
                              hipStream_t stream) {
  const float* content     = (const float*)d_in[0];
  const float* style       = (const float*)d_in[1];
  const float* content_key = (const float*)d_in[2];
  const float* style_key   = (const float*)d_in[3];
  const float* Wf = (const float*)d_in[4];
  const float* bf = (const float*)d_in[5];
  const float* Wg = (const float*)d_in[6];
  const float* bg = (const float*)d_in[7];
  const float* Wh = (const float*)d_in[8];
  const float* bh = (const float*)d_in[9];

  float* out = (float*)d_out;
  float* S   = out + (long)BATCH * CCH * NC;    // outputs: (out, S) concatenated

  float* ws = (float*)d_ws;
  size_t off = 0;
  float* style_tmp = ws + off; off += (size_t)BATCH * CCH * 128 * 64;
  float* style_r   = ws + off; off += (size_t)BATCH * CCH * 64 * 64;
  float* Fq        = ws + off; off += (size_t)BATCH * QKC * NC;
  float* Gk        = ws + off; off += (size_t)BATCH * QKC * NS;
  float* Hr        = ws + off; off += (size_t)BATCH * CCH * NS;
  float* Hr2       = ws + off; off += (size_t)BATCH * CCH * NS;
  float* Lg        = ws + off; off += (size_t)BATCH * NC * NS;
  float* rmax      = ws + off; off += (size_t)BATCH * NC;
  float* rinv      = ws + off; off += (size_t)BATCH * NC;
  float* cmean     = ws + off; off += (size_t)BATCH * CCH;
  float* cinv      = ws + off; off += (size_t)BATCH * CCH;

  resize_h<<<(BATCH * CCH * 128 * 64) / 256, 256, 0, stream>>>(style, style_tmp);
  resize_v<<<(BATCH * CCH * 64 * 64) / 256, 256, 0, stream>>>(style_tmp, style_r);

  conv1x1_wmma<<<dim3(NC / 64, QKC / 128, BATCH), 256, 0, stream>>>(
      Wf, content_key, bf, Fq, nullptr, QKC, QKC, NC);
  conv1x1_wmma<<<dim3(NS / 64, QKC / 128, BATCH), 256, 0, stream>>>(
      Wg, style_key, bg, Gk, nullptr, QKC, QKC, NS);
  conv1x1_wmma<<<dim3(NS / 64, CCH / 128, BATCH), 256, 0, stream>>>(
      Wh, style_r, bh, Hr, Hr2, CCH, CCH, NS);

  attn_logits_wmma<<<dim3(NS / 64, NC / 128, BATCH), 256, 0, stream>>>(Fq, Gk, Lg);

  row_softmax_stats<<<(BATCH * NC) / 4, 128, 0, stream>>>(Lg, rmax, rinv);
  content_stats<<<(BATCH * CCH) / 4, 128, 0, stream>>>(content, cmean, cinv);

  attn_apply_wmma<<<dim3(NC / 16, BATCH), 256, 0, stream>>>(
      Lg, rmax, rinv, Hr, Hr2, content, cmean, cinv, out);

  write_S<<<((size_t)BATCH * NC * NS) / 256, 256, 0, stream>>>(Lg, rmax, rinv, S);
}